// BiDirectionalAttention_63642825392568
// MI455X (gfx1250) — compile-verified
//
#include <hip/hip_runtime.h>
#include <hip/hip_bf16.h>
#include <cstdint>

#define B_ 32
#define C_ 2048
#define Q_ 64
#define H_ 128
#define NEGV (-1000000000.0f)

typedef float v2f __attribute__((ext_vector_type(2)));
typedef float v8f __attribute__((ext_vector_type(8)));

// D = A(16x4 f32) * B(4x16 f32) + C(16x16 f32), full f32 precision on tensor path.
__device__ __forceinline__ v8f wmma_f32(v2f a, v2f b, v8f c) {
  return __builtin_amdgcn_wmma_f32_16x16x4_f32(false, a, false, b, (short)0, c,
                                               false, false);
}

// ---------------------------------------------------------------------------
// Kernel 1: per (b, 64 ctx rows). Each wave owns a 16-row tile.
//   sim = (ctx .* w_m) @ question^T + ctx.w_c + question.w_q  (WMMA GEMM 1)
//   softmax over q  ->  att ;  rowmax -> workspace
//   q2c = att @ question                                      (WMMA GEMM 2)
//   out[:,:,0:H]=ctx  out[:,:,H:2H]=q2c  out[:,:,2H:3H]=ctx*q2c
// ---------------------------------------------------------------------------
__global__ __launch_bounds__(128, 1) void bidaf_main(
    const float* __restrict__ ctx, const float* __restrict__ qst,
    const unsigned char* __restrict__ qmask, const float* __restrict__ wgt,
    float* __restrict__ out, float* __restrict__ rowm) {
  __shared__ float q_lds[Q_][132];        // question tile, padded stride
  __shared__ float att_lds[4][16][68];    // per-wave att tile (C->A relayout)
  __shared__ float qb_lds[Q_];            // question . w_q
  __shared__ float wc_lds[H_];
  __shared__ float wm_lds[H_];
  __shared__ float qm_lds[Q_];            // 1.0 => masked out
  __shared__ float cd_lds[4][32];         // partial ctx . w_c

  const int b    = blockIdx.y;
  const int tid  = threadIdx.x;
  const int wave = tid >> 5;
  const int lane = tid & 31;
  const int half = lane >> 4;   // K-half selector per WMMA layout
  const int ln   = lane & 15;
  const int c0   = blockIdx.x * 64 + wave * 16;

  // Stage question tile (64x128 f32 = 32KB) coalesced into LDS.
  const float* qbase = qst + (size_t)b * Q_ * H_;
  for (int i = tid; i < Q_ * (H_ / 4); i += 128) {
    const int r  = i >> 5;
    const int c4 = (i & 31) << 2;
    *(float4*)(&q_lds[r][c4]) = *(const float4*)(qbase + r * H_ + c4);
  }
  if (tid < H_) {
    wc_lds[tid] = wgt[tid];
    wm_lds[tid] = wgt[2 * H_ + tid];
  }
  if (tid < Q_) qm_lds[tid] = qmask[b * Q_ + tid] ? 0.f : 1.f;
  __syncthreads();
  if (tid < Q_) {  // qbias[q] = question[q,:] . w_q
    float s = 0.f;
    for (int k = 0; k < H_; ++k) s = fmaf(q_lds[tid][k], wgt[H_ + k], s);
    qb_lds[tid] = s;
  }
  __syncthreads();

  // ---- GEMM 1: 16x64 sim tile via 32 K-steps x 4 q-tiles of WMMA ----
  const v8f vzero = {};
  v8f acc[4];
#pragma unroll
  for (int t = 0; t < 4; ++t) acc[t] = vzero;

  const float* aptr = ctx + ((size_t)b * C_ + c0 + ln) * H_ + half * 2;
  float pc = 0.f;  // partial ctx.w_c for this lane's (row, K-half) slice
#pragma unroll 4
  for (int k0 = 0; k0 < H_; k0 += 4) {
    const int kb   = k0 + half * 2;
    const float ax = aptr[k0];
    const float ay = aptr[k0 + 1];
    pc = fmaf(ax, wc_lds[kb], fmaf(ay, wc_lds[kb + 1], pc));
    v2f a;
    a.x = ax * wm_lds[kb];
    a.y = ay * wm_lds[kb + 1];
#pragma unroll
    for (int t = 0; t < 4; ++t) {
      v2f bf;  // B[k][n] = question[n0+n][k]
      bf.x   = q_lds[t * 16 + ln][kb];
      bf.y   = q_lds[t * 16 + ln][kb + 1];
      acc[t] = wmma_f32(a, bf, acc[t]);
    }
  }
  cd_lds[wave][lane] = pc;
  __syncthreads();

  // ---- biases + mask + masked softmax over q (wave shuffles) ----
  float mx[8], sm[8];
#pragma unroll
  for (int v = 0; v < 8; ++v) {
    const int M    = v + half * 8;  // row within 16-row tile (C/D layout)
    const float cd = cd_lds[wave][M] + cd_lds[wave][M + 16];
    float m = -__builtin_inff();
#pragma unroll
    for (int t = 0; t < 4; ++t) {
      float s = acc[t][v] + cd + qb_lds[t * 16 + ln];
      if (qm_lds[t * 16 + ln] != 0.f) s = NEGV;
      acc[t][v] = s;
      m = fmaxf(m, s);
    }
#pragma unroll
    for (int off = 1; off < 16; off <<= 1)
      m = fmaxf(m, __shfl_xor(m, off, 32));
    mx[v] = m;
    float ssum = 0.f;
#pragma unroll
    for (int t = 0; t < 4; ++t) {
      const float ev = __expf(acc[t][v] - m);
      acc[t][v]      = ev;
      ssum += ev;
    }
#pragma unroll
    for (int off = 1; off < 16; off <<= 1) ssum += __shfl_xor(ssum, off, 32);
    sm[v] = ssum;
  }
  if (ln == 0) {  // rowmax -> workspace for the global c2q softmax
#pragma unroll
    for (int v = 0; v < 8; ++v)
      rowm[(size_t)b * C_ + c0 + v + half * 8] = mx[v];
  }
  // normalized att -> per-wave LDS tile (relayout C-frag -> A-frag)
#pragma unroll
  for (int v = 0; v < 8; ++v) {
    const float inv = 1.0f / sm[v];
    const int M     = v + half * 8;
#pragma unroll
    for (int t = 0; t < 4; ++t)
      att_lds[wave][M][t * 16 + ln] = acc[t][v] * inv;
  }

  // ---- GEMM 2: q2c(16x128) = att(16x64) @ question(64x128) ----
  v8f oacc[8];
#pragma unroll
  for (int t = 0; t < 8; ++t) oacc[t] = vzero;
#pragma unroll 2
  for (int k0 = 0; k0 < Q_; k0 += 4) {
    const int kb = k0 + half * 2;
    v2f a;
    a.x = att_lds[wave][ln][kb];
    a.y = att_lds[wave][ln][kb + 1];
#pragma unroll
    for (int t = 0; t < 8; ++t) {
      v2f bf;  // B[k][n] = question[q=k][h=n]
      bf.x    = q_lds[kb][t * 16 + ln];
      bf.y    = q_lds[kb + 1][t * 16 + ln];
      oacc[t] = wmma_f32(a, bf, oacc[t]);
    }
  }

  // ---- epilogue: first three H-slices of fused output ----
#pragma unroll
  for (int v = 0; v < 8; ++v) {
    const int c       = c0 + v + half * 8;
    const float* crow = ctx + ((size_t)b * C_ + c) * H_;
    float* orow       = out + ((size_t)b * C_ + c) * (4 * H_);
#pragma unroll
    for (int t = 0; t < 8; ++t) {
      const int h    = t * 16 + ln;
      const float cv = crow[h];
      const float qv = oacc[t][v];
      orow[h]            = cv;
      orow[H_ + h]       = qv;
      orow[2 * H_ + h]   = cv * qv;
    }
  }
}

// ---------------------------------------------------------------------------
// Kernel 2: per batch b — softmax over 2048 rowmaxes, then c2q[b,:] (tiny).
// ---------------------------------------------------------------------------
__global__ __launch_bounds__(256, 1) void bidaf_c2q(
    const float* __restrict__ ctx, const float* __restrict__ rowm,
    float* __restrict__ c2q) {
  __shared__ float red[256];
  __shared__ float att_s[C_];
  const int b   = blockIdx.x;
  const int tid = threadIdx.x;

  float m = -__builtin_inff();
  for (int c = tid; c < C_; c += 256)
    m = fmaxf(m, rowm[(size_t)b * C_ + c]);
  red[tid] = m;
  __syncthreads();
  for (int s = 128; s > 0; s >>= 1) {
    if (tid < s) red[tid] = fmaxf(red[tid], red[tid + s]);
    __syncthreads();
  }
  const float mxv = red[0];
  __syncthreads();

  float sum = 0.f;
  for (int c = tid; c < C_; c += 256) {
    const float e = __expf(rowm[(size_t)b * C_ + c] - mxv);
    att_s[c]      = e;
    sum += e;
  }
  red[tid] = sum;
  __syncthreads();
  for (int s = 128; s > 0; s >>= 1) {
    if (tid < s) red[tid] += red[tid + s];
    __syncthreads();
  }
  const float inv = 1.0f / red[0];
  __syncthreads();

  const int h     = tid & (H_ - 1);
  const int chunk = tid >> 7;  // two C-chunks of 1024
  float a         = 0.f;
  const float* cb = ctx + (size_t)b * C_ * H_;
  for (int c = chunk * (C_ / 2); c < (chunk + 1) * (C_ / 2); ++c)
    a = fmaf(att_s[c], cb[(size_t)c * H_ + h], a);
  red[tid] = a;
  __syncthreads();
  if (tid < H_) c2q[b * H_ + tid] = (red[tid] + red[tid + H_]) * inv;
}

// ---------------------------------------------------------------------------
// Kernel 3: out[:,:,3H:4H] = ctx * c2q[b]  (float4, fully coalesced)
// ---------------------------------------------------------------------------
__global__ __launch_bounds__(256, 1) void bidaf_out3(
    const float* __restrict__ ctx, const float* __restrict__ c2q,
    float* __restrict__ out) {
  const size_t idx = (size_t)blockIdx.x * 256 + threadIdx.x;  // float4 index
  const int h4     = (int)(idx & (H_ / 4 - 1));
  const size_t bc  = idx >> 5;        // H/4 == 32
  const int b      = (int)(bc >> 11); // C == 2048
  const float4 cv  = ((const float4*)ctx)[idx];
  const float4 sv  = ((const float4*)c2q)[(size_t)b * (H_ / 4) + h4];
  float4 r;
  r.x = cv.x * sv.x;
  r.y = cv.y * sv.y;
  r.z = cv.z * sv.z;
  r.w = cv.w * sv.w;
  ((float4*)out)[bc * (4 * H_ / 4) + 3 * (H_ / 4) + h4] = r;
}

extern "C" void kernel_launch(void* const* d_in, const int* in_sizes, int n_in,
                              void* d_out, int out_size, void* d_ws,
                              size_t ws_size, hipStream_t stream) {
  (void)in_sizes; (void)n_in; (void)out_size; (void)ws_size;
  const float* ctx            = (const float*)d_in[0];
  const float* qst            = (const float*)d_in[1];
  // d_in[2] = context_mask: unused by the reference computation
  const unsigned char* qmask  = (const unsigned char*)d_in[3];
  const float* wgt            = (const float*)d_in[4];
  float* out  = (float*)d_out;
  float* rowm = (float*)d_ws;               // B*C floats
  float* c2q  = rowm + (size_t)B_ * C_;     // B*H floats

  dim3 g1(C_ / 64, B_);
  bidaf_main<<<g1, 128, 0, stream>>>(ctx, qst, qmask, wgt, out, rowm);
  bidaf_c2q<<<B_, 256, 0, stream>>>(ctx, rowm, c2q);
  bidaf_out3<<<(B_ * C_ * (H_ / 4)) / 256, 256, 0, stream>>>(ctx, c2q, out);
}